// PointNetSaModuleKNN_11888469475433
// MI455X (gfx1250) — compile-verified
//
#include <hip/hip_runtime.h>
#include <hip/hip_bf16.h>

// ---------------------------------------------------------------------------
// PointNet++ SA module (KNN variant) for MI455X / gfx1250, wave32.
// Pipeline: fps -> pack -> knn -> [gemm0(WMMA f32)+stats] -> bnfin ->
//           [gemm1(TDM stage + WMMA)+stats] -> bnfin ->
//           [gemm2(WMMA)+bias+max -> out]
// ---------------------------------------------------------------------------

#define BATCH   8
#define NPTS    16384
#define NPOINT  2048
#define KNN_K   32
#define FEAT    64
#define CIN     67
#define CIN_P   68                     // padded K (row 67 zeroed)
#define MTOT    (BATCH*NPOINT*KNN_K)   // 524288 rows
#define MTILE   128
#define NBLK    (MTOT/MTILE)           // 4096 blocks

typedef float v2f __attribute__((ext_vector_type(2)));
typedef float v8f __attribute__((ext_vector_type(8)));
typedef unsigned int u32x4 __attribute__((ext_vector_type(4)));
typedef int i32x4 __attribute__((ext_vector_type(4)));
typedef int i32x8 __attribute__((ext_vector_type(8)));

#if defined(__has_builtin)
#  if __has_builtin(__builtin_amdgcn_tensor_load_to_lds) && \
      __has_builtin(__builtin_amdgcn_s_wait_tensorcnt)
#    define USE_TDM 1
#  endif
#endif
#ifndef USE_TDM
#  define USE_TDM 0
#endif

// ---------------------------------------------------------------------------
// 1) Furthest point sampling: one block per batch; dist state in registers.
//    Writes new_xyz (B,3,NPOINT) directly to the output buffer.
// ---------------------------------------------------------------------------
__global__ __launch_bounds__(1024) void fps_kernel(const float* __restrict__ xyz,
                                                   float* __restrict__ new_xyz) {
  const int b = blockIdx.x;
  const int t = threadIdx.x;
  const float* xb = xyz + (size_t)b * 3 * NPTS;
  const int PER = NPTS / 1024;  // 16 points per thread

  float px[16], py[16], pz[16], dist[16];
#pragma unroll
  for (int i = 0; i < PER; i++) {
    int j = t + i * 1024;
    px[i] = xb[j];
    py[i] = xb[NPTS + j];
    pz[i] = xb[2 * NPTS + j];
    dist[i] = 1e10f;
  }

  __shared__ float s_val[32];
  __shared__ int   s_idx[32];
  __shared__ int   s_far;
  __shared__ float s_c[3];

  int far = 0;
  for (int k = 0; k < NPOINT; k++) {
    if (t < 3) {
      float v = xb[t * NPTS + far];
      new_xyz[((size_t)b * 3 + t) * NPOINT + k] = v;
      s_c[t] = v;
    }
    __syncthreads();
    float cx = s_c[0], cy = s_c[1], cz = s_c[2];

    float best = -1.0f; int bi = 0;
#pragma unroll
    for (int i = 0; i < PER; i++) {
      float dx = px[i] - cx, dy = py[i] - cy, dz = pz[i] - cz;
      float d = dx * dx + dy * dy + dz * dz;
      float nd = fminf(dist[i], d);
      dist[i] = nd;
      if (nd > best) { best = nd; bi = t + i * 1024; }
    }
    // wave32 argmax (tie -> lower index)
#pragma unroll
    for (int off = 16; off > 0; off >>= 1) {
      float ov = __shfl_xor(best, off, 32);
      int   oi = __shfl_xor(bi,   off, 32);
      if (ov > best || (ov == best && oi < bi)) { best = ov; bi = oi; }
    }
    int wid = t >> 5;
    if ((t & 31) == 0) { s_val[wid] = best; s_idx[wid] = bi; }
    __syncthreads();
    if (t < 32) {
      best = s_val[t]; bi = s_idx[t];
#pragma unroll
      for (int off = 16; off > 0; off >>= 1) {
        float ov = __shfl_xor(best, off, 32);
        int   oi = __shfl_xor(bi,   off, 32);
        if (ov > best || (ov == best && oi < bi)) { best = ov; bi = oi; }
      }
      if (t == 0) s_far = bi;
    }
    __syncthreads();
    far = s_far;
  }
}

// ---------------------------------------------------------------------------
// 2) Pack per-point features as (B, N, 68): [xyz(3) | points(64) | 0-pad]
//    so the GEMM0 gather reads one contiguous 272B row per sample.
// ---------------------------------------------------------------------------
__global__ __launch_bounds__(256) void pack_kernel(const float* __restrict__ xyz,
                                                   const float* __restrict__ points,
                                                   float* __restrict__ ptf) {
  const int b = blockIdx.x >> 6;             // NPTS/256 = 64 blocks per batch
  const int n = ((blockIdx.x & 63) << 8) + threadIdx.x;
  float* dst = ptf + ((size_t)b * NPTS + n) * CIN_P;
#pragma unroll
  for (int c = 0; c < 3; c++)
    dst[c] = xyz[((size_t)b * 3 + c) * NPTS + n];
  for (int c = 0; c < FEAT; c++)
    dst[3 + c] = points[((size_t)b * FEAT + c) * NPTS + n];
  dst[CIN] = 0.0f;
}

// ---------------------------------------------------------------------------
// 3) KNN: one wave per query. Each lane builds a sorted 32-entry candidate
//    list in LDS over its N/32 strided points, then a 32-round wave argmin
//    merge produces the global top-32 (ties -> lower index).
// ---------------------------------------------------------------------------
#define KNN_WPB 4
__global__ __launch_bounds__(128) void knn_kernel(const float* __restrict__ xyz,
                                                  const float* __restrict__ new_xyz,
                                                  int* __restrict__ kidx) {
  __shared__ float sd_all[KNN_WPB * 32 * 33];
  __shared__ int   si_all[KNN_WPB * 32 * 33];
  const int lane = threadIdx.x & 31;
  const int wv = threadIdx.x >> 5;
  const int q = blockIdx.x * KNN_WPB + wv;   // [0, B*NPOINT)
  const int b = q >> 11;
  const int p = q & 2047;

  const float* xb = xyz + (size_t)b * 3 * NPTS;
  const float qx = new_xyz[((size_t)b * 3 + 0) * NPOINT + p];
  const float qy = new_xyz[((size_t)b * 3 + 1) * NPOINT + p];
  const float qz = new_xyz[((size_t)b * 3 + 2) * NPOINT + p];

  float* sd = sd_all + (wv * 32 + lane) * 33;
  int*   si = si_all + (wv * 32 + lane) * 33;

  int cnt = 0;
  float worst = 3.0e38f;
  for (int j = lane; j < NPTS; j += 32) {
    float dx = xb[j] - qx;
    float dy = xb[NPTS + j] - qy;
    float dz = xb[2 * NPTS + j] - qz;
    float d = dx * dx + dy * dy + dz * dz;
    if (cnt < KNN_K || d < worst) {
      int pos = (cnt < KNN_K) ? cnt : (KNN_K - 1);
      while (pos > 0 && sd[pos - 1] > d) {
        sd[pos] = sd[pos - 1];
        si[pos] = si[pos - 1];
        pos--;
      }
      sd[pos] = d;
      si[pos] = j;
      if (cnt < KNN_K) cnt++;
      if (cnt == KNN_K) worst = sd[KNN_K - 1];
    }
  }
  // wave-wide 32-round merge
  int ptr = 0;
  for (int r = 0; r < KNN_K; r++) {
    float v  = (ptr < cnt) ? sd[ptr] : 3.0e38f;
    int   id = (ptr < cnt) ? si[ptr] : 0x7FFFFFFF;
    float bv = v; int bid = id;
#pragma unroll
    for (int off = 16; off > 0; off >>= 1) {
      float ov = __shfl_xor(bv,  off, 32);
      int   oi = __shfl_xor(bid, off, 32);
      if (ov < bv || (ov == bv && oi < bid)) { bv = ov; bid = oi; }
    }
    if (v == bv && id == bid) ptr++;   // indices unique -> exactly one lane
    if (lane == 0) kidx[(size_t)q * KNN_K + r] = bid;
  }
}

// ---------------------------------------------------------------------------
// 4) GEMM0 (gather + W0 + b0), WMMA f32 16x16x4. Also emits deterministic
//    per-block BN partial stats {sum, sumsq} per channel.
// ---------------------------------------------------------------------------
#define AST0 132
__global__ __launch_bounds__(256) void gemm0_kernel(const float* __restrict__ ptf,
                                                    const float* __restrict__ new_xyz,
                                                    const int*   __restrict__ kidx,
                                                    const float* __restrict__ W0,
                                                    const float* __restrict__ b0,
                                                    float* __restrict__ h0,
                                                    float* __restrict__ part) {
  __shared__ float a_tile[CIN_P * AST0];   // [c][m], padded stride
  __shared__ float b_tile[CIN_P * 64];     // [c][o]
  __shared__ float s_sum[8 * 64];
  __shared__ float s_sq[8 * 64];
  const int t = threadIdx.x;
  const int rowBase = blockIdx.x * MTILE;

  // stage W0^T (zero-pad K row 67)
  for (int i = t; i < CIN_P * 64; i += 256) {
    int c = i >> 6, o = i & 63;
    b_tile[i] = (c < CIN) ? W0[o * CIN + c] : 0.0f;
  }
  // stage gathered A tile: 128 rows x 68 channels (K-major)
  {
    const int r = t & 127;
    const int half = t >> 7;
    const int gr = rowBase + r;
    const int b = gr >> 16;              // NPOINT*KNN_K = 65536 rows/batch
    const int rem = gr & 65535;
    const int p = rem >> 5;
    const int s = rem & 31;
    const int j = kidx[((b * NPOINT + p) << 5) + s];
    const float* src = ptf + ((size_t)b * NPTS + j) * CIN_P;
    const float* nx = new_xyz + (size_t)b * 3 * NPOINT + p;
#pragma unroll
    for (int cc = 0; cc < 34; cc++) {
      int c = half * 34 + cc;
      float v = src[c];
      if (c < 3) v -= nx[(size_t)c * NPOINT];
      if (c >= CIN) v = 0.0f;
      a_tile[c * AST0 + r] = v;
    }
  }
  __syncthreads();

  const int lane = t & 31;
  const int wv = t >> 5;
  const int m = lane & 15;
  const int klo = (lane >> 4) * 2;

  v8f acc[4] = {};
#pragma unroll
  for (int k0 = 0; k0 < CIN_P; k0 += 4) {
    v2f a;
    a.x = a_tile[(k0 + klo)     * AST0 + wv * 16 + m];
    a.y = a_tile[(k0 + klo + 1) * AST0 + wv * 16 + m];
#pragma unroll
    for (int nt = 0; nt < 4; nt++) {
      v2f bb;
      bb.x = b_tile[(k0 + klo)     * 64 + nt * 16 + m];
      bb.y = b_tile[(k0 + klo + 1) * 64 + nt * 16 + m];
      acc[nt] = __builtin_amdgcn_wmma_f32_16x16x4_f32(false, a, false, bb,
                                                      (short)0, acc[nt],
                                                      false, false);
    }
  }

  // bias, store, per-wave BN partials
  const int rowHalf = (lane >> 4) * 8;
#pragma unroll
  for (int nt = 0; nt < 4; nt++) {
    const int col = nt * 16 + m;
    const float bias = b0[col];
    float S = 0.0f, Q = 0.0f;
#pragma unroll
    for (int i = 0; i < 8; i++) {
      float v = acc[nt][i] + bias;
      h0[(size_t)(rowBase + wv * 16 + rowHalf + i) * 64 + col] = v;
      S += v; Q += v * v;
    }
    S += __shfl_xor(S, 16, 32);
    Q += __shfl_xor(Q, 16, 32);
    if (lane < 16) { s_sum[wv * 64 + col] = S; s_sq[wv * 64 + col] = Q; }
  }
  __syncthreads();
  if (t < 64) {
    float S = 0.0f, Q = 0.0f;
#pragma unroll
    for (int w = 0; w < 8; w++) { S += s_sum[w * 64 + t]; Q += s_sq[w * 64 + t]; }
    part[blockIdx.x * 128 + t] = S;
    part[blockIdx.x * 128 + 64 + t] = Q;
  }
}

// ---------------------------------------------------------------------------
// 5) BN finalize: fold mean/var/gamma/beta into per-channel scale/shift.
// ---------------------------------------------------------------------------
__global__ __launch_bounds__(64) void bnfin_kernel(const float* __restrict__ part,
                                                   const float* __restrict__ g,
                                                   const float* __restrict__ bt,
                                                   float* __restrict__ bn) {
  const int c = threadIdx.x;
  float S = 0.0f, Q = 0.0f;
  for (int blk = 0; blk < NBLK; blk++) {
    S += part[blk * 128 + c];
    Q += part[blk * 128 + 64 + c];
  }
  const float inv = 1.0f / (float)MTOT;
  const float mean = S * inv;
  const float var = Q * inv - mean * mean;
  const float sc = g[c] * rsqrtf(var + 1e-5f);
  bn[c] = sc;
  bn[64 + c] = bt[c] - mean * sc;
}

// ---------------------------------------------------------------------------
// 6) GEMM1: stage raw 128x64 h0 tile to LDS via the Tensor Data Mover (one
//    TENSOR_LOAD_TO_LDS per block; D# pad feature gives a 68-float row
//    stride -> conflict-free A-fragment ds_load_b64). BN0+ReLU is applied at
//    fragment-load time. Falls back to a synchronous copy into the same
//    padded layout if the TDM builtin is unavailable.
// ---------------------------------------------------------------------------
#define RST1 68   // padded row stride (floats): 64 + 4 (TDM pad_amount)
__global__ __launch_bounds__(256) void gemm1_kernel(const float* __restrict__ h0,
                                                    const float* __restrict__ bn0,
                                                    const float* __restrict__ W1,
                                                    const float* __restrict__ b1,
                                                    float* __restrict__ h1,
                                                    float* __restrict__ part) {
  __shared__ float a_tile[MTILE * RST1];   // [r][c] raw h0, padded rows
  __shared__ float b_tile[64 * 64];        // [c][o]
  __shared__ float s_sum[8 * 64];
  __shared__ float s_sq[8 * 64];
  __shared__ float s_bn[128];
  const int t = threadIdx.x;
  const int lane = t & 31;
  const int wv = t >> 5;
  const int rowBase = blockIdx.x * MTILE;

  if (t < 128) s_bn[t] = bn0[t];
  for (int i = t; i < 64 * 64; i += 256) {
    int c = i >> 6, o = i & 63;
    b_tile[i] = W1[o * 64 + c];
  }

#if USE_TDM
  if (wv == 0) {
    // D# per CDNA5 ISA: 2-D tensor, 4B elements,
    // tile 64 x 128, row stride 64, LDS pad: +4 DW every 64 DW (stride 68).
    unsigned long long ga = (unsigned long long)(size_t)(h0 + (size_t)rowBase * 64);
    unsigned int lds = (unsigned int)(size_t)(&a_tile[0]);
    u32x4 g0;
    g0[0] = 1u;                                        // count=1 (valid)
    g0[1] = lds;                                       // lds_addr
    g0[2] = (unsigned int)(ga & 0xFFFFFFFFu);          // global_addr[31:0]
    g0[3] = (unsigned int)((ga >> 32) & 0x01FFFFFFu)   // global_addr[56:32]
            | (2u << 30);                              // type=2 ("image")
    i32x8 g1;
    g1[0] = (2 << 16)          // data_size=4B
            | (1 << 20)        // pad_enable
            | (5 << 22)        // pad_interval: 64 DWORDs
            | (3 << 25);       // pad_amount: 4 DWORDs
    g1[1] = (64 << 16);        // tensor_dim0 = 64
    g1[2] = (128 << 16);       // tensor_dim1 = 128
    g1[3] = (64 << 16);        // tile_dim0 = 64
    g1[4] = 128;               // tile_dim1 = 128
    g1[5] = 64;                // tensor_dim0_stride = 64
    g1[6] = 0;
    g1[7] = 0;
    i32x4 gz4 = {0, 0, 0, 0};
#if defined(__clang_major__) && (__clang_major__ >= 23)
    i32x8 gz8 = {0, 0, 0, 0, 0, 0, 0, 0};
    __builtin_amdgcn_tensor_load_to_lds(g0, g1, gz4, gz4, gz8, 0);
#else
    __builtin_amdgcn_tensor_load_to_lds(g0, g1, gz4, gz4, 0);
#endif
    __builtin_amdgcn_s_wait_tensorcnt((short)0);
  }
#else
  // fallback: synchronous staged copy into the same padded layout
  for (int u = t; u < MTILE * 16; u += 256) {  // 16B chunks
    int r = u >> 4, ch = u & 15;
    const float4 v = ((const float4*)(h0 + (size_t)(rowBase + r) * 64))[ch];
    a_tile[r * RST1 + ch * 4 + 0] = v.x;
    a_tile[r * RST1 + ch * 4 + 1] = v.y;
    a_tile[r * RST1 + ch * 4 + 2] = v.z;
    a_tile[r * RST1 + ch * 4 + 3] = v.w;
  }
#endif
  __syncthreads();

  const int m = lane & 15;
  const int klo = (lane >> 4) * 2;
  const int arow = (wv * 16 + m) * RST1;

  v8f acc[4] = {};
#pragma unroll
  for (int k0 = 0; k0 < 64; k0 += 4) {
    const int c0 = k0 + klo;
    // BN0 + ReLU applied at fragment load
    v2f a;
    a.x = fmaxf(a_tile[arow + c0]     * s_bn[c0]     + s_bn[64 + c0],     0.0f);
    a.y = fmaxf(a_tile[arow + c0 + 1] * s_bn[c0 + 1] + s_bn[64 + c0 + 1], 0.0f);
#pragma unroll
    for (int nt = 0; nt < 4; nt++) {
      v2f bb;
      bb.x = b_tile[(c0)     * 64 + nt * 16 + m];
      bb.y = b_tile[(c0 + 1) * 64 + nt * 16 + m];
      acc[nt] = __builtin_amdgcn_wmma_f32_16x16x4_f32(false, a, false, bb,
                                                      (short)0, acc[nt],
                                                      false, false);
    }
  }

  const int rowHalf = (lane >> 4) * 8;
#pragma unroll
  for (int nt = 0; nt < 4; nt++) {
    const int col = nt * 16 + m;
    const float bias = b1[col];
    float S = 0.0f, Q = 0.0f;
#pragma unroll
    for (int i = 0; i < 8; i++) {
      float v = acc[nt][i] + bias;
      h1[(size_t)(rowBase + wv * 16 + rowHalf + i) * 64 + col] = v;
      S += v; Q += v * v;
    }
    S += __shfl_xor(S, 16, 32);
    Q += __shfl_xor(Q, 16, 32);
    if (lane < 16) { s_sum[wv * 64 + col] = S; s_sq[wv * 64 + col] = Q; }
  }
  __syncthreads();
  if (t < 64) {
    float S = 0.0f, Q = 0.0f;
#pragma unroll
    for (int w = 0; w < 8; w++) { S += s_sum[w * 64 + t]; Q += s_sq[w * 64 + t]; }
    part[blockIdx.x * 128 + t] = S;
    part[blockIdx.x * 128 + 64 + t] = Q;
  }
}

// ---------------------------------------------------------------------------
// 7) GEMM2: fused BN1+ReLU on load, WMMA (N=128), bias, max over k=32,
//    writes new_points (B,128,NPOINT). Each 128-row tile = 4 full s-groups.
// ---------------------------------------------------------------------------
__global__ __launch_bounds__(256) void gemm2_kernel(const float* __restrict__ h1,
                                                    const float* __restrict__ bn1,
                                                    const float* __restrict__ W2,
                                                    const float* __restrict__ b2,
                                                    float* __restrict__ out_np) {
  __shared__ float smem[64 * 128 + 64 * 128];   // a_tile | b_tile (64KB)
  float* a_tile = smem;            // [c][m], stride 128
  float* b_tile = smem + 64 * 128; // [c][o], o in [0,128)
  const int t = threadIdx.x;
  const int rowBase = blockIdx.x * MTILE;

  for (int i = t; i < 64 * 128; i += 256) {
    int c = i >> 7, o = i & 127;
    b_tile[i] = W2[o * 64 + c];
  }
  {
    const int r = t >> 1;
    const int c0 = (t & 1) * 32;
    const float* src = h1 + (size_t)(rowBase + r) * 64;
#pragma unroll
    for (int cc = 0; cc < 32; cc++) {
      int c = c0 + cc;
      float v = fmaxf(src[c] * bn1[c] + bn1[64 + c], 0.0f);
      a_tile[c * 128 + r] = v;
    }
  }
  __syncthreads();

  const int lane = t & 31;
  const int wv = t >> 5;
  const int m = lane & 15;
  const int klo = (lane >> 4) * 2;

  v8f acc[8] = {};
#pragma unroll
  for (int k0 = 0; k0 < 64; k0 += 4) {
    v2f a;
    a.x = a_tile[(k0 + klo)     * 128 + wv * 16 + m];
    a.y = a_tile[(k0 + klo + 1) * 128 + wv * 16 + m];
#pragma unroll
    for (int nt = 0; nt < 8; nt++) {
      v2f bb;
      bb.x = b_tile[(k0 + klo)     * 128 + nt * 16 + m];
      bb.y = b_tile[(k0 + klo + 1) * 128 + nt * 16 + m];
      acc[nt] = __builtin_amdgcn_wmma_f32_16x16x4_f32(false, a, false, bb,
                                                      (short)0, acc[nt],
                                                      false, false);
    }
  }
  __syncthreads();                  // done with a_tile -> alias as s_max
  float* s_max = smem;              // [8 waves][128 ch]

#pragma unroll
  for (int nt = 0; nt < 8; nt++) {
    float mv = acc[nt][0];
#pragma unroll
    for (int i = 1; i < 8; i++) mv = fmaxf(mv, acc[nt][i]);
    mv = fmaxf(mv, __shfl_xor(mv, 16, 32));   // both 8-row halves of the wave
    if (lane < 16) s_max[wv * 128 + nt * 16 + m] = mv;
  }
  __syncthreads();

  // combine wave pairs (each p-group = 32 rows = 2 waves), add bias, store
  for (int u = t; u < 512; u += 256) {
    const int pg = u >> 7;
    const int ch = u & 127;
    float v = fmaxf(s_max[(2 * pg) * 128 + ch], s_max[(2 * pg + 1) * 128 + ch]) + b2[ch];
    const int gr = rowBase + pg * 32;
    const int b = gr >> 16;
    const int p = (gr & 65535) >> 5;
    out_np[((size_t)b * 128 + ch) * NPOINT + p] = v;
  }
}

// ---------------------------------------------------------------------------
// Launch
// ---------------------------------------------------------------------------
extern "C" void kernel_launch(void* const* d_in, const int* in_sizes, int n_in,
                              void* d_out, int out_size, void* d_ws, size_t ws_size,
                              hipStream_t stream) {
  const float* xyz    = (const float*)d_in[0];
  const float* points = (const float*)d_in[1];
  const float* W0  = (const float*)d_in[2];
  const float* b0  = (const float*)d_in[3];
  const float* g0  = (const float*)d_in[4];
  const float* bt0 = (const float*)d_in[5];
  const float* W1  = (const float*)d_in[6];
  const float* b1  = (const float*)d_in[7];
  const float* g1  = (const float*)d_in[8];
  const float* bt1 = (const float*)d_in[9];
  const float* W2  = (const float*)d_in[10];
  const float* b2  = (const float*)d_in[11];

  float* out_newxyz = (float*)d_out;                                  // (B,3,NPOINT)
  float* out_newpts = (float*)d_out + (size_t)BATCH * 3 * NPOINT;     // (B,128,NPOINT)

  char* ws = (char*)d_ws;
  const size_t sz_ptf = (size_t)BATCH * NPTS * CIN_P * sizeof(float);
  const size_t sz_idx = (size_t)BATCH * NPOINT * KNN_K * sizeof(int);
  const size_t sz_h   = (size_t)MTOT * 64 * sizeof(float);
  const size_t sz_prt = (size_t)NBLK * 128 * sizeof(float);

  float* ptf   = (float*)ws;                       ws += sz_ptf;
  int*   kidx  = (int*)ws;                         ws += sz_idx;
  float* h0    = (float*)ws;                       ws += sz_h;
  float* h1    = (float*)ws;                       ws += sz_h;
  float* part0 = (float*)ws;                       ws += sz_prt;
  float* part1 = (float*)ws;                       ws += sz_prt;
  float* bn0   = (float*)ws;                       ws += 128 * sizeof(float);
  float* bn1   = (float*)ws;

  fps_kernel<<<BATCH, 1024, 0, stream>>>(xyz, out_newxyz);
  pack_kernel<<<BATCH * (NPTS / 256), 256, 0, stream>>>(xyz, points, ptf);
  knn_kernel<<<(BATCH * NPOINT) / KNN_WPB, 128, 0, stream>>>(xyz, out_newxyz, kidx);
  gemm0_kernel<<<NBLK, 256, 0, stream>>>(ptf, out_newxyz, kidx, W0, b0, h0, part0);
  bnfin_kernel<<<1, 64, 0, stream>>>(part0, g0, bt0, bn0);
  gemm1_kernel<<<NBLK, 256, 0, stream>>>(h0, bn0, W1, b1, h1, part1);
  bnfin_kernel<<<1, 64, 0, stream>>>(part1, g1, bt1, bn1);
  gemm2_kernel<<<NBLK, 256, 0, stream>>>(h1, bn1, W2, b2, out_newpts);
}